// LSTMModel_84018150244648
// MI455X (gfx1250) — compile-verified
//
#include <hip/hip_runtime.h>
#include <math.h>

typedef __attribute__((ext_vector_type(2))) float v2f;
typedef __attribute__((ext_vector_type(4))) float v4f;
typedef __attribute__((ext_vector_type(8))) float v8f;

#define B_SZ 512
#define T_SZ 512
#define I_SZ 128
#define H_SZ 50
#define G_SZ 200   // 4*H
#define NT   13    // gate tiles: 13*16 = 208 >= 200
#define BBLK 16    // batch rows per scan block
#define NBT  (B_SZ / BBLK)   // 32 batch tiles
#define O_SZ 10

__device__ __forceinline__ float sigmoidf_(float x) {
    return 1.0f / (1.0f + __expf(-x));
}

// -------------------------------------------------------------------------
// Kernel 1: x_proj = x @ W_ih^T + (b_ih + b_hh), stored in WMMA-fragment
// tile layout: xp[((t*NBT + btile)*NT + ntile)*256 + lane*8 + r], where
// (lane, r) follows the 16x16 f32 C/D layout (row = 8*(lane/16)+r, col =
// lane%16). One 16x16 tile per wave; K=128 via 32 WMMA in 2 chains.
// -------------------------------------------------------------------------
__global__ __launch_bounds__(256) void xproj_wmma(
    const float* __restrict__ x, const float* __restrict__ W_ih,
    const float* __restrict__ b_ih, const float* __restrict__ b_hh,
    float* __restrict__ xp)
{
    const int lane   = threadIdx.x & 31;
    const int wid    = threadIdx.x >> 5;
    const int tile   = blockIdx.x * 8 + wid;       // = m_tile*NT + n_tile
    const int m_tile = tile / NT;
    const int n_tile = tile - m_tile * NT;
    const int m0     = m_tile << 4;
    const int n0     = n_tile << 4;

    // 16 rows of this tile share one t (512 % 16 == 0)
    const int t = m0 >> 9;                  // m0 / 512
    const int b = (m0 & 511) + (lane & 15);
    const float* arow = x + ((size_t)b * T_SZ + t) * I_SZ;

    const int n  = n0 + (lane & 15);
    const int nc = (n < G_SZ) ? n : (G_SZ - 1);   // clamp; cols >=200 unused later
    const float* brow = W_ih + (size_t)nc * I_SZ;
    const int koff = (lane >> 4) << 1;            // 0 or 2

    v8f acc0 = {};
    v8f acc1 = {};
#pragma unroll
    for (int k0 = 0; k0 < I_SZ; k0 += 8) {        // two independent WMMA chains
        v2f a0 = *(const v2f*)(arow + k0 + koff);
        v2f b0 = *(const v2f*)(brow + k0 + koff);
        acc0 = __builtin_amdgcn_wmma_f32_16x16x4_f32(false, a0, false, b0,
                                                     (short)0, acc0, false, false);
        v2f a1 = *(const v2f*)(arow + k0 + 4 + koff);
        v2f b1 = *(const v2f*)(brow + k0 + 4 + koff);
        acc1 = __builtin_amdgcn_wmma_f32_16x16x4_f32(false, a1, false, b1,
                                                     (short)0, acc1, false, false);
    }

    const float bias = (n < G_SZ) ? (b_ih[n] + b_hh[n]) : 0.0f;

    float* dst = xp + (size_t)tile * 256 + lane * 8;   // 32B-aligned, coalesced
    v4f lo, hi;
#pragma unroll
    for (int r = 0; r < 4; ++r) lo[r] = acc0[r] + acc1[r] + bias;
#pragma unroll
    for (int r = 0; r < 4; ++r) hi[r] = acc0[4 + r] + acc1[4 + r] + bias;
    *(v4f*)(dst)     = lo;   // global_store_b128
    *(v4f*)(dst + 4) = hi;   // global_store_b128
}

// -------------------------------------------------------------------------
// Kernel 2: recurrent scan. Each block owns BBLK=16 batch rows (independent
// across blocks => no inter-block sync). 13 waves; wave w computes gate
// columns [16w, 16w+16). W_hh B-fragments live in registers for the whole
// kernel; h/c in LDS. Per step: gates = x_proj_tile + h @ W_hh^T via 13
// WMMA (2 chains), then elementwise LSTM cell. Epilogue: h @ W_fc^T + b_fc.
// -------------------------------------------------------------------------
__global__ __launch_bounds__(416) void lstm_scan_wmma(
    const float* __restrict__ xp, const float* __restrict__ W_hh,
    const float* __restrict__ W_fc, const float* __restrict__ b_fc,
    float* __restrict__ out)
{
    __shared__ float h_lds[BBLK][52];       // K padded 50 -> 52, pad stays 0
    __shared__ float c_lds[BBLK][52];
    __shared__ float g_lds[BBLK][NT * 16];  // 208 cols

    const int tid  = threadIdx.x;
    const int lane = tid & 31;
    const int wid  = tid >> 5;              // 0..12 (N-tile of this wave)
    const int bt   = blockIdx.x;            // batch tile 0..31
    const int b0   = bt * BBLK;

    // ---- preload W_hh fragments: B[k][n] = W_hh[n][k], K padded to 52 ----
    const int n = (wid << 4) + (lane & 15); // global gate column, < 208
    v2f bfrag[13];
#pragma unroll
    for (int ks = 0; ks < 13; ++ks) {
        const int k = (ks << 2) + ((lane >> 4) << 1);
        float c0 = 0.0f, c1 = 0.0f;
        if (n < G_SZ) {
            if (k < H_SZ)     c0 = W_hh[(size_t)n * H_SZ + k];
            if (k + 1 < H_SZ) c1 = W_hh[(size_t)n * H_SZ + k + 1];
        }
        bfrag[ks].x = c0;
        bfrag[ks].y = c1;
    }

    // ---- zero-init h, c (including K padding columns) ----
    for (int i = tid; i < BBLK * 52; i += blockDim.x) {
        (&h_lds[0][0])[i] = 0.0f;
        (&c_lds[0][0])[i] = 0.0f;
    }
    __syncthreads();

    const int mrow  = lane & 15;
    const int koff  = (lane >> 4) << 1;
    const int rbase = (lane >> 4) << 3;

    for (int t = 0; t < T_SZ; ++t) {
        // C0 init = x_proj tile in fragment layout: 2 coalesced b128 loads
        const float* src = xp + ((size_t)(t * NBT + bt) * NT + wid) * 256 + lane * 8;
        v4f lo = *(const v4f*)(src);
        v4f hi = *(const v4f*)(src + 4);
        v8f acc0;
#pragma unroll
        for (int r = 0; r < 4; ++r) { acc0[r] = lo[r]; acc0[4 + r] = hi[r]; }
        v8f acc1 = {};

        if (t + 1 < T_SZ) {  // prefetch next step's tile -> global_prefetch_b8
            const float* nxt = xp + ((size_t)((t + 1) * NBT + bt) * NT + wid) * 256
                                  + lane * 8;
            __builtin_prefetch(nxt, 0, 3);
        }

        // gates += h @ W_hh^T : 13 WMMA split into 2 independent chains
#pragma unroll
        for (int ks = 0; ks < 13; ks += 2) {
            v2f a = *(const v2f*)(&h_lds[mrow][(ks << 2) + koff]);
            acc0 = __builtin_amdgcn_wmma_f32_16x16x4_f32(false, a, false, bfrag[ks],
                                                         (short)0, acc0, false, false);
        }
#pragma unroll
        for (int ks = 1; ks < 13; ks += 2) {
            v2f a = *(const v2f*)(&h_lds[mrow][(ks << 2) + koff]);
            acc1 = __builtin_amdgcn_wmma_f32_16x16x4_f32(false, a, false, bfrag[ks],
                                                         (short)0, acc1, false, false);
        }

        // spill gate tile to LDS (row-major [16][208])
#pragma unroll
        for (int r = 0; r < 8; ++r)
            g_lds[rbase + r][n] = acc0[r] + acc1[r];
        __syncthreads();

        // elementwise LSTM cell (PyTorch gate order i,f,g,o)
        for (int item = tid; item < BBLK * H_SZ; item += blockDim.x) {
            const int bi = item / H_SZ;
            const int j  = item - bi * H_SZ;
            const float ig = sigmoidf_(g_lds[bi][j]);
            const float fg = sigmoidf_(g_lds[bi][H_SZ + j]);
            const float gg = tanhf(g_lds[bi][2 * H_SZ + j]);
            const float og = sigmoidf_(g_lds[bi][3 * H_SZ + j]);
            const float cn = fg * c_lds[bi][j] + ig * gg;
            c_lds[bi][j] = cn;
            h_lds[bi][j] = og * tanhf(cn);
        }
        __syncthreads();
    }

    // ---- final FC: out[b][o] = h[b] . W_fc[o] + b_fc[o] ----
    for (int item = tid; item < BBLK * O_SZ; item += blockDim.x) {
        const int bi = item / O_SZ;
        const int o  = item - bi * O_SZ;
        float s = b_fc[o];
#pragma unroll
        for (int j = 0; j < H_SZ; ++j)
            s += h_lds[bi][j] * W_fc[o * H_SZ + j];
        out[(size_t)(b0 + bi) * O_SZ + o] = s;
    }
}

// -------------------------------------------------------------------------
extern "C" void kernel_launch(void* const* d_in, const int* in_sizes, int n_in,
                              void* d_out, int out_size, void* d_ws, size_t ws_size,
                              hipStream_t stream) {
    (void)in_sizes; (void)n_in; (void)out_size; (void)ws_size;
    const float* x    = (const float*)d_in[0];
    const float* W_ih = (const float*)d_in[1];
    const float* W_hh = (const float*)d_in[2];
    const float* b_ih = (const float*)d_in[3];
    const float* b_hh = (const float*)d_in[4];
    const float* W_fc = (const float*)d_in[5];
    const float* b_fc = (const float*)d_in[6];
    float* out = (float*)d_out;
    float* xp  = (float*)d_ws;   // fragment-layout x_proj: 16384*13*256*4 B ~ 218 MB

    // Kernel 1: 16384 M-tiles x 13 N-tiles, 8 wave-tiles per 256-thread block
    const int tiles = (B_SZ * T_SZ / 16) * NT;     // 212992
    xproj_wmma<<<tiles / 8, 256, 0, stream>>>(x, W_ih, b_ih, b_hh, xp);

    // Kernel 2: 32 independent batch-slices, 13 waves each
    lstm_scan_wmma<<<B_SZ / BBLK, 13 * 32, 0, stream>>>(xp, W_hh, W_fc, b_fc, out);
}